// CTCLayer_16819091931368
// MI455X (gfx1250) — compile-verified
//
#include <hip/hip_runtime.h>
#include <stdint.h>

#define NEGF (-1.0e30f)

// 3-way logsumexp; numerically identical to nested jnp.logaddexp for these inputs
__device__ __forceinline__ float lse3(float a, float b, float c) {
    float m = fmaxf(fmaxf(a, b), c);
    float s = expf(a - m) + expf(b - m) + expf(c - m);
    return m + logf(s);
}

__device__ __forceinline__ float lae(float x, float y) {
    float mx = fmaxf(x, y);
    float mn = fminf(x, y);
    return mx + log1pf(expf(mn - mx));
}

// gfx1250 async global->LDS 16B copy, GVS mode: SGPR64 base + per-lane u32 byte offset.
// Tracked with ASYNCcnt (CDNA5 ISA).
__device__ __forceinline__ void async_load_b128(void* lds_ptr, const float* base, uint32_t byte_off) {
    uint32_t lds_addr = (uint32_t)(uintptr_t)lds_ptr;  // low 32 bits of generic ptr == LDS offset
    asm volatile("global_load_async_to_lds_b128 %0, %1, %2"
                 :: "v"(lds_addr), "v"(byte_off), "s"(base)
                 : "memory");
}

__device__ __forceinline__ void wait_asynccnt0() {
    asm volatile("s_wait_asynccnt 0x0" ::: "memory");
}

__global__ __launch_bounds__(96)
void ctc_batch_cost_kernel(const float* __restrict__ y_pred,
                           const int*   __restrict__ labels,
                           const int*   __restrict__ input_length,
                           const int*   __restrict__ label_length,
                           float*       __restrict__ out,
                           int T, int C, int U)
{
    const int b     = blockIdx.x;
    const int tid   = threadIdx.x;
    const int blank = C - 1;
    const int S     = 2 * U + 1;            // 65 for U=32
    const int NV    = C / 4;                // 25 b128 loader lanes (C multiple of 4)

    __shared__ __align__(16) float rowbuf[2][128];  // double-buffered softmax row (C<=128)
    __shared__ float               alpha[2][80];    // double-buffered DP state (S<=80)

    const int llen = label_length[b];
    int       ilen = input_length[b];
    if (ilen > T) ilen = T;
    const int sValid = 2 * llen + 1;
    const int tEnd   = (ilen < 1) ? 1 : ilen;       // updates happen for t in [1, ilen)

    const float* base = y_pred + (size_t)b * (size_t)T * (size_t)C;

    // ---- kick off async load of row t=0 (wave 0 only) ----
    if (tid < NV) {
        async_load_b128(&rowbuf[0][tid * 4], base, (uint32_t)(tid * 16));
    }

    // extended labels in registers (loop-invariant):
    // blank, l0, blank, l1, ..., blank ; skip iff odd state & label != previous label
    int  myExt   = blank;
    bool mySkip  = false;
    bool myValid = false;
    if (tid < S) {
        myValid = (tid < sValid);
        if (tid & 1) {
            int li = tid >> 1;
            myExt  = labels[b * U + li];
            int pv = (li > 0) ? labels[b * U + li - 1] : -1;
            mySkip = (tid >= 2) && (myExt != pv);
        }
    }

    wait_asynccnt0();
    __syncthreads();   // row 0 landed across all waves

    // ---- init alpha at t=0 into buffer 0 ----
    if (tid < S) {
        float a = NEGF;
        if (tid <= 1) {
            float e = logf(rowbuf[0][myExt] + 1e-7f);
            a = (tid == 0) ? e : ((llen > 0) ? e : NEGF);
        }
        if (!myValid) a = NEGF;
        alpha[0][tid] = a;
    }

    // prefetch row t=1 into buffer 1
    if (tEnd > 1 && tid < NV) {
        async_load_b128(&rowbuf[1][tid * 4], base, (uint32_t)(C * 4 + tid * 16));
    }

    // ---- forward recursion: one barrier per step ----
    for (int t = 1; t < tEnd; ++t) {
        const int cur = t & 1;   // row buffer & alpha write buffer
        const int rd  = cur ^ 1; // alpha read buffer

        wait_asynccnt0();     // wave 0: its row-t slices are in LDS (no-op for other waves)
        __syncthreads();      // row[cur] + alpha[rd] writes visible everywhere;
                              // all reads of rowbuf[rd] / alpha[cur] from step t-1 retired

        // prefetch row t+1 into the buffer consumed at step t-1
        if (t + 1 < tEnd && tid < NV) {
            async_load_b128(&rowbuf[cur ^ 1][tid * 4], base,
                            (uint32_t)((t + 1) * C * 4 + tid * 16));
        }

        if (tid < S) {
            float e  = logf(rowbuf[cur][myExt] + 1e-7f);
            float a0 = alpha[rd][tid];
            float a1 = (tid >= 1) ? alpha[rd][tid - 1] : NEGF;
            float a2 = mySkip ? alpha[rd][tid - 2] : NEGF;
            float m  = lse3(a0, a1, a2) + e;
            alpha[cur][tid] = myValid ? m : NEGF;
        }
    }

    __syncthreads();
    if (tid == 0) {
        const int fin  = (tEnd - 1) & 1;   // buffer holding the final alpha
        int   last = 2 * llen;
        float aL   = alpha[fin][last];
        float aP   = (llen > 0) ? alpha[fin][(last > 0) ? last - 1 : 0] : NEGF;
        out[b] = -lae(aL, aP);
    }
}

extern "C" void kernel_launch(void* const* d_in, const int* in_sizes, int n_in,
                              void* d_out, int out_size, void* d_ws, size_t ws_size,
                              hipStream_t stream) {
    const float* y_pred = (const float*)d_in[0];
    const int*   labels = (const int*)d_in[1];
    const int*   ilen   = (const int*)d_in[2];
    const int*   llen   = (const int*)d_in[3];
    float*       outp   = (float*)d_out;

    const int B  = in_sizes[2];            // one length per batch element
    const int U  = in_sizes[1] / B;        // 32
    const int C  = 100;                    // classes per reference
    const int T  = (in_sizes[0] / B) / C;  // 512

    ctc_batch_cost_kernel<<<B, 96, 0, stream>>>(y_pred, labels, ilen, llen, outp, T, C, U);
}